// ContinuousNormalizingFlow_29901562314989
// MI455X (gfx1250) — compile-verified
//
#include <hip/hip_runtime.h>

typedef float v2f __attribute__((ext_vector_type(2)));
typedef float v8f __attribute__((ext_vector_type(8)));

namespace {
constexpr int   kDim    = 16;
constexpr int   kHid    = 256;
constexpr int   kBatch  = 8192;
constexpr int   kSteps  = 10;
constexpr int   kPitch  = 260;   // W2T row pitch (floats): conflict-free ds_load_b64
constexpr int   kBlock  = 256;   // 8 waves per WG (LDS limits to 1 WG/WGP -> 2 waves/SIMD)
constexpr int   kGrid   = 256;   // 2048 waves -> 4 batch elements per wave (uniform)
constexpr float kLog2Pi = 1.8378770664093453f;
}

__global__ __launch_bounds__(kBlock, 1)
void cnf_logp_kernel(const float* __restrict__ x,
                     const float* __restrict__ W1, const float* __restrict__ b1,
                     const float* __restrict__ W2, const float* __restrict__ b2,
                     const float* __restrict__ W3, const float* __restrict__ b3,
                     float* __restrict__ out)
{
    // ---- LDS: 266240+16384+17408+16384+512+1024+1024+64 = 319,040 B (< 320 KB) ----
    __shared__ __align__(16) float w2t[kHid * kPitch];     // W2 transposed [n][k], padded
    __shared__ __align__(16) float waA[64 * 64];           // W1 rows 0..15 in WMMA A-layout
    __shared__ __align__(16) float w1s[(kDim + 1) * kHid]; // W1 row-major (for layer-1 fwd)
    __shared__ __align__(16) float squad[8 * 512];         // per-wave (s1,s1,h1,h1) quads
    __shared__ __align__(16) float fzsc[8 * 16];           // per-wave fz bounce buffer
    __shared__ __align__(16) float b1s[kHid];
    __shared__ __align__(16) float b2s[kHid];
    __shared__ __align__(16) float b3s[kDim];

    const int tid  = threadIdx.x;
    const int lane = tid & 31;
    const int wv   = tid >> 5;
    const int nlo  = lane & 15;   // WMMA B/C lane -> column N
    const int kh   = lane >> 4;   // lane half -> K sub-offset (A/B) / M half (C)

    // ---- prologue: stage all weights into LDS (one-time, L2-backed) ----
    for (int idx = tid; idx < kHid * kHid; idx += kBlock) {
        const int k = idx >> 8, n = idx & 255;
        w2t[n * kPitch + k] = W2[idx];
    }
    for (int idx = tid; idx < (kDim + 1) * kHid; idx += kBlock) w1s[idx] = W1[idx];
    for (int idx = tid; idx < kHid; idx += kBlock) { b1s[idx] = b1[idx]; b2s[idx] = b2[idx]; }
    if (tid < kDim) b3s[tid] = b3[tid];
    // W1 rows 0..15 -> A-matrix layout: lane l holds W1[l&15, 4j + 2*(l>>4) + v]
    if (wv == 0) {
        for (int j = 0; j < 64; ++j) {
            const float* p = W1 + nlo * kHid + 4 * j + 2 * kh;
            waA[j * 64 + lane * 2 + 0] = p[0];
            waA[j * 64 + lane * 2 + 1] = p[1];
        }
    }
    __syncthreads();

    float* sq  = squad + wv * 512;
    float* fzv = fzsc  + wv * 16;

    const float dt = 1.0f / (float)kSteps;
    const int gw = blockIdx.x * 8 + wv;
    const int nw = kGrid * 8;

    for (int b = gw; b < kBatch; b += nw) {
        float zc   = (lane < kDim) ? x[b * kDim + lane] : 0.0f;  // lane l holds z[l]
        float lacc = 0.0f;                                       // logp1 accumulator

        for (int step = 0; step < kSteps; ++step) {
            const float t0 = (float)step * dt;
            float kz = 0.0f;                 // previous stage dz/dt (per-lane)
            float zsum = 0.0f, dsum = 0.0f;  // RK4 weighted sums

            for (int s = 0; s < 4; ++s) {
                const float a  = (s == 0) ? 0.0f : ((s == 3) ? 1.0f : 0.5f);
                const float wr = (s == 1 || s == 2) ? 2.0f : 1.0f;
                const float ts = t0 + a * dt;
                const float zs = zc + (a * dt) * kz;

                // broadcast stage-input z to all lanes
                float zb[kDim];
                #pragma unroll
                for (int k = 0; k < kDim; ++k) zb[k] = __shfl(zs, k, 32);

                // ---------- layer 1: u1 = [z,t]@W1 + b1 ; stash silu/silu' ----------
                #pragma unroll
                for (int q = 0; q < 8; ++q) {
                    const int n = lane + 32 * q;
                    float u = b1s[n];
                    #pragma unroll
                    for (int k = 0; k < kDim; ++k) u = fmaf(zb[k], w1s[k * kHid + n], u);
                    u = fmaf(ts, w1s[kDim * kHid + n], u);
                    const float sg = 1.0f / (1.0f + __expf(-u));
                    sq[(n >> 1) * 4 + (n & 1)]     = sg * (1.0f + u * (1.0f - sg)); // silu'
                    sq[(n >> 1) * 4 + 2 + (n & 1)] = u * sg;                        // silu
                }
                asm volatile("s_wait_dscnt 0" ::: "memory");

                // ---------- tangent GEMM (16x256 @ 256x256) fused with u2 forward ----------
                v8f   acc[16];   // du2 tiles, WMMA C-layout
                float up[16];    // u2 partials in B-layout
                #pragma unroll
                for (int nt = 0; nt < 16; ++nt) {
                    v8f z0;
                    #pragma unroll
                    for (int e = 0; e < 8; ++e) z0[e] = 0.0f;
                    acc[nt] = z0;
                    up[nt]  = 0.0f;
                }

                #pragma unroll 2
                for (int j = 0; j < 64; ++j) {
                    const int k0 = 4 * j;
                    // quad: {s1[k0+2kh], s1[k0+2kh+1], h1[k0+2kh], h1[k0+2kh+1]}
                    const float4 sh = *(const float4*)&sq[(2 * j + kh) * 4];
                    const v2f    wa = *(const v2f*)&waA[j * 64 + lane * 2];
                    v2f A;
                    A.x = wa.x * sh.x;
                    A.y = wa.y * sh.y;
                    #pragma unroll
                    for (int nt = 0; nt < 16; ++nt) {
                        const v2f Bt = *(const v2f*)&w2t[(nt * 16 + nlo) * kPitch + k0 + 2 * kh];
                        acc[nt] = __builtin_amdgcn_wmma_f32_16x16x4_f32(
                            false, A, false, Bt, (short)0, acc[nt], false, false);
                        up[nt] = fmaf(sh.z, Bt.x, fmaf(sh.w, Bt.y, up[nt]));
                    }
                }

                // ---------- layer 2 activation, layer 3 forward + divergence ----------
                float divp = 0.0f;
                float fzp[8];
                #pragma unroll
                for (int v = 0; v < 8; ++v) fzp[v] = 0.0f;

                #pragma unroll
                for (int nt = 0; nt < 16; ++nt) {
                    const int n = nt * 16 + nlo;
                    const float u2 = up[nt] + __shfl_xor(up[nt], 16, 32) + b2s[n];
                    const float sg = 1.0f / (1.0f + __expf(-u2));
                    const float h2 = u2 * sg;
                    const float s2 = sg * (1.0f + u2 * (1.0f - sg));
                    const float4* w3p = (const float4*)(W3 + n * kDim + 8 * kh);
                    const float4 wa3 = w3p[0];
                    const float4 wb3 = w3p[1];
                    const v8f d = acc[nt] * s2;  // dh2 rows m = 8*kh + v, column n
                    divp += d[0]*wa3.x + d[1]*wa3.y + d[2]*wa3.z + d[3]*wa3.w
                          + d[4]*wb3.x + d[5]*wb3.y + d[6]*wb3.z + d[7]*wb3.w;
                    fzp[0] = fmaf(h2, wa3.x, fzp[0]);
                    fzp[1] = fmaf(h2, wa3.y, fzp[1]);
                    fzp[2] = fmaf(h2, wa3.z, fzp[2]);
                    fzp[3] = fmaf(h2, wa3.w, fzp[3]);
                    fzp[4] = fmaf(h2, wb3.x, fzp[4]);
                    fzp[5] = fmaf(h2, wb3.y, fzp[5]);
                    fzp[6] = fmaf(h2, wb3.z, fzp[6]);
                    fzp[7] = fmaf(h2, wb3.w, fzp[7]);
                }

                // wave reductions
                #pragma unroll
                for (int m = 1; m <= 16; m <<= 1) divp += __shfl_xor(divp, m, 32);
                #pragma unroll
                for (int v = 0; v < 8; ++v) {
                    #pragma unroll
                    for (int m = 1; m <= 8; m <<= 1) fzp[v] += __shfl_xor(fzp[v], m, 32);
                }
                if (nlo == 0) {  // lanes 0 and 16 hold fz[0..7] / fz[8..15]
                    float4 lo = {fzp[0], fzp[1], fzp[2], fzp[3]};
                    float4 hi = {fzp[4], fzp[5], fzp[6], fzp[7]};
                    *(float4*)&fzv[8 * kh + 0] = lo;
                    *(float4*)&fzv[8 * kh + 4] = hi;
                }
                asm volatile("s_wait_dscnt 0" ::: "memory");
                kz = (lane < kDim) ? (fzv[lane] + b3s[lane]) : 0.0f;

                zsum += wr * kz;
                dsum += wr * divp;
            }

            zc   += (dt / 6.0f) * zsum;
            lacc += (dt / 6.0f) * (-dsum);   // dlogp/dt = -div
        }

        // log p(x) = logN(z1) - logp1
        float qq = zc * zc;
        #pragma unroll
        for (int m = 1; m <= 16; m <<= 1) qq += __shfl_xor(qq, m, 32);
        if (lane == 0) {
            out[b] = -0.5f * (qq + (float)kDim * kLog2Pi) - lacc;
        }
    }
}

extern "C" void kernel_launch(void* const* d_in, const int* in_sizes, int n_in,
                              void* d_out, int out_size, void* d_ws, size_t ws_size,
                              hipStream_t stream)
{
    (void)in_sizes; (void)n_in; (void)out_size; (void)d_ws; (void)ws_size;
    const float* x  = (const float*)d_in[0];
    const float* W1 = (const float*)d_in[1];
    const float* b1 = (const float*)d_in[2];
    const float* W2 = (const float*)d_in[3];
    const float* b2 = (const float*)d_in[4];
    const float* W3 = (const float*)d_in[5];
    const float* b3 = (const float*)d_in[6];
    cnf_logp_kernel<<<kGrid, kBlock, 0, stream>>>(x, W1, b1, W2, b2, W3, b3, (float*)d_out);
}